// Clusterer_27401891348969
// MI455X (gfx1250) — compile-verified
//
#include <hip/hip_runtime.h>
#include <math.h>

#define NROWS   262144
#define DDIM    16
#define KDIM    64
#define TAU     0.1f
#define LOG2PI  1.8378770664093453f

#define AGRID   128
#define ABLK    256
#define CGRID   512
#define CBLK    256            // 8 waves of 32
#define TILES   (NROWS / 16)   // 16384
#define TOTWAVES (CGRID * CBLK / 32)  // 4096

typedef __attribute__((ext_vector_type(2))) float v2f;
typedef __attribute__((ext_vector_type(8))) float v8f;

// ---------------------------------------------------------------------------
// Kernel A: per-dim min/max partials of met_locs (for R = max - min)
// ---------------------------------------------------------------------------
__global__ void kminmax(const float* __restrict__ met,
                        float* __restrict__ pmin, float* __restrict__ pmax) {
    __shared__ float smin[ABLK], smax[ABLK];
    const int t = threadIdx.x;          // dim class = t & 15 (stride 256*128 % 16 == 0)
    const size_t total  = (size_t)NROWS * DDIM;
    const size_t stride = (size_t)AGRID * ABLK;
    float mn = 3.4e38f, mx = -3.4e38f;
    for (size_t i = (size_t)blockIdx.x * ABLK + t; i < total; i += stride) {
        float v = met[i];
        mn = fminf(mn, v); mx = fmaxf(mx, v);
    }
    smin[t] = mn; smax[t] = mx;
    __syncthreads();
    for (int off = ABLK / 2; off >= DDIM; off >>= 1) {
        if (t < off) {
            smin[t] = fminf(smin[t], smin[t + off]);
            smax[t] = fmaxf(smax[t], smax[t + off]);
        }
        __syncthreads();
    }
    if (t < DDIM) {
        pmin[blockIdx.x * DDIM + t] = smin[t];
        pmax[blockIdx.x * DDIM + t] = smax[t];
    }
}

// ---------------------------------------------------------------------------
// Kernel B (1 block, 64 threads): per-k constants + all small losses
// wsK[k] = { logpi[k], -0.5*exp(-r[k]), -0.5*D*(r[k]+log2pi), ||mu_k||^2 }
// wsScal[0] = const0 = lgamma(K) + (K-1)*log(tau) + sum(logpi)
// wsScal[1] = pi_loss + mu_loss + lambda_loss + b_loss + r_loss + C_loss
// ---------------------------------------------------------------------------
__global__ void kprep(const float* __restrict__ mu, const float* __restrict__ pi,
                      const float* __restrict__ lambda_mu, const float* __restrict__ b,
                      const float* __restrict__ Cv, const float* __restrict__ r,
                      const float* __restrict__ pmin, const float* __restrict__ pmax,
                      float* __restrict__ wsScal, float4* __restrict__ wsK) {
    __shared__ float sR[DDIM];
    __shared__ float sred[KDIM];
    __shared__ float sconst[4];  // 0:G 1:c 2:g 3:sum(log var_mu)
    const int t = threadIdx.x;   // 0..63

    if (t < DDIM) {
        float mn = 3.4e38f, mx = -3.4e38f;
        for (int i = 0; i < AGRID; ++i) {
            mn = fminf(mn, pmin[i * DDIM + t]);
            mx = fmaxf(mx, pmax[i * DDIM + t]);
        }
        sR[t] = mx - mn;
    }
    __syncthreads();
    if (t == 0) {
        float s2 = 0.f;
        for (int d = 0; d < DDIM; ++d) s2 += sR[d] * sR[d];
        const float c = 1.25f + (DDIM - 1) / 4.0f;   // 5.0
        const float g = 0.25f + (DDIM - 1) / 4.0f;   // 4.0
        sconst[0] = c / (50.f * g) * sqrtf(s2);      // G
        sconst[1] = c;
        sconst[2] = g;
        float slv = 0.f;
        for (int d = 0; d < DDIM; ++d)
            slv += logf(lambda_mu[d] * lambda_mu[d] * sR[d]);
        sconst[3] = slv;
    }
    __syncthreads();

    // log_softmax over pi (K=64)
    const float piv = pi[t];
    sred[t] = piv; __syncthreads();
    for (int off = 32; off > 0; off >>= 1) {
        if (t < off) sred[t] = fmaxf(sred[t], sred[t + off]);
        __syncthreads();
    }
    const float pimax = sred[0]; __syncthreads();
    sred[t] = expf(piv - pimax); __syncthreads();
    for (int off = 32; off > 0; off >>= 1) {
        if (t < off) sred[t] += sred[t + off];
        __syncthreads();
    }
    const float logpi = piv - (pimax + logf(sred[0])); __syncthreads();
    sred[t] = logpi; __syncthreads();
    for (int off = 32; off > 0; off >>= 1) {
        if (t < off) sred[t] += sred[t + off];
        __syncthreads();
    }
    const float sum_logpi = sred[0]; __syncthreads();

    // per-k work
    const float G = sconst[0], c = sconst[1], g = sconst[2], slogvar = sconst[3];
    float mu2 = 0.f, quad = 0.f, bsq = 0.f;
    for (int d = 0; d < DDIM; ++d) {
        const float m   = mu[t * DDIM + d];
        const float bv  = b[t * DDIM + d];
        const float lam = lambda_mu[d];
        const float var = lam * lam * sR[d];
        const float dmb = m - bv;
        mu2  += m * m;
        quad += dmb * dmb / var;
        bsq  += bv * bv;
    }
    const float rk = r[t], Ck = Cv[t];
    wsK[t] = make_float4(logpi, -0.5f * expf(-rk),
                         -0.5f * (float)DDIM * (rk + LOG2PI), mu2);

    float part = 0.f;
    part += -((1.0f / KDIM - 1.0f) * logpi);                                 // pi_loss
    part += 0.5f * quad + 0.5f * slogvar + 0.5f * DDIM * LOG2PI;             // mu_loss
    part += 0.5f * bsq;                                                      // b_loss quad
    part += -(c * logf(Ck) - (c - 1.0f) * rk - Ck * expf(-rk) - lgammaf(c)); // r_loss
    part += -(g * logf(G)  - (g - 1.0f) * Ck - G * expf(-Ck) - lgammaf(g));  // C_loss
    if (t < DDIM) {
        const float lam = lambda_mu[t];
        part += -(0.5f * logf(0.5f) - lgammaf(0.5f) - 0.5f * lam - 0.5f * expf(lam)); // lambda
    }
    if (t == 0) part += 0.5f * (float)KDIM * (float)DDIM * LOG2PI;           // b_loss const

    sred[t] = part; __syncthreads();
    for (int off = 32; off > 0; off >>= 1) {
        if (t < off) sred[t] += sred[t + off];
        __syncthreads();
    }
    if (t == 0) {
        wsScal[0] = lgammaf((float)KDIM) + (KDIM - 1) * logf(TAU) + sum_logpi;
        wsScal[1] = sred[0];
    }
}

// ---------------------------------------------------------------------------
// Kernel C: one wave per 16-row tile. WMMA f32 16x16x4 chained over D=16,
// 4 column tiles cover K=64. Row reductions via half-wave shfl_xor.
// ---------------------------------------------------------------------------
__global__ void __launch_bounds__(CBLK)
kmain(const float* __restrict__ met, const float* __restrict__ mu,
      const float* __restrict__ z,
      const float* __restrict__ wsScal, const float4* __restrict__ wsK,
      float* __restrict__ wavePart) {
    const int lane = threadIdx.x & 31;
    const int half = lane >> 4;
    const int l16  = lane & 15;
    const int wgid = blockIdx.x * (CBLK / 32) + (threadIdx.x >> 5);

    const float const0 = wsScal[0];

    // per-lane cluster constants; k = 16c + l16 (same for both halves)
    float lp[4], kB[4], kC[4], m2[4];
    for (int c = 0; c < 4; ++c) {
        const float4 kk = wsK[16 * c + l16];
        lp[c] = kk.x; kB[c] = kk.y; kC[c] = kk.z; m2[c] = kk.w;
    }

    // B fragments (mu^T): B VGPR0 holds K'=2*half, VGPR1 K'=2*half+1; column N=l16
    v2f bf[4][4];
    for (int c = 0; c < 4; ++c) {
        const float* mrow = mu + (16 * c + l16) * DDIM;
        for (int j = 0; j < 4; ++j) {
            bf[c][j].x = mrow[4 * j + 2 * half];
            bf[c][j].y = mrow[4 * j + 2 * half + 1];
        }
    }

    float wsum = 0.f;
    for (int tile = wgid; tile < TILES; tile += TOTWAVES) {
        const int m0 = tile << 4;
        // A fragments: row M=l16; VGPR0 K'=2*half, VGPR1 K'=2*half+1
        const float* xrow = met + (size_t)(m0 + l16) * DDIM;
        v2f afrag[4];
        float p2 = 0.f;
        for (int j = 0; j < 4; ++j) {
            const float x0 = xrow[4 * j + 2 * half];
            const float x1 = xrow[4 * j + 2 * half + 1];
            afrag[j].x = x0; afrag[j].y = x1;
            p2 += x0 * x0 + x1 * x1;
        }
        // ||x_row(l16)||^2 (both halves together hold all 16 dims)
        const float x2row = p2 + __shfl_xor(p2, 16, 32);
        float x2m[8];  // x2 of row (v + 8*half), fetched within own half
        for (int v = 0; v < 8; ++v) x2m[v] = __shfl(x2row, v + 8 * half, 16);

        float zf[4][8], av[4][8];
        for (int c = 0; c < 4; ++c) {
            v8f acc = {0.f, 0.f, 0.f, 0.f, 0.f, 0.f, 0.f, 0.f};
            for (int j = 0; j < 4; ++j)
                acc = __builtin_amdgcn_wmma_f32_16x16x4_f32(
                    false, afrag[j], false, bf[c][j], (short)0, acc, false, false);
            // C/D layout: VGPR v -> M = v + 8*half, N = l16
            for (int v = 0; v < 8; ++v) {
                const int m = m0 + v + 8 * half;
                const float zkv  = z[(size_t)m * KDIM + 16 * c + l16];
                const float sq   = x2m[v] - 2.0f * acc[v] + m2[c];
                const float logN = kB[c] * sq + kC[c];
                zf[c][v] = zkv;
                av[c][v] = zkv + logN;
            }
        }

        for (int v = 0; v < 8; ++v) {
            // lse_z, sum_z over k
            float zm = fmaxf(fmaxf(zf[0][v], zf[1][v]), fmaxf(zf[2][v], zf[3][v]));
            for (int m = 1; m < 16; m <<= 1) zm = fmaxf(zm, __shfl_xor(zm, m, 16));
            float se = 0.f, sz = 0.f;
            for (int c = 0; c < 4; ++c) { se += expf(zf[c][v] - zm); sz += zf[c][v]; }
            for (int m = 1; m < 16; m <<= 1) {
                se += __shfl_xor(se, m, 16); sz += __shfl_xor(sz, m, 16);
            }
            const float lse_z = zm + logf(se);
            // lse_a = logsumexp(z + logN)
            float am = fmaxf(fmaxf(av[0][v], av[1][v]), fmaxf(av[2][v], av[3][v]));
            for (int m = 1; m < 16; m <<= 1) am = fmaxf(am, __shfl_xor(am, m, 16));
            float sa = 0.f;
            for (int c = 0; c < 4; ++c) sa += expf(av[c][v] - am);
            for (int m = 1; m < 16; m <<= 1) sa += __shfl_xor(sa, m, 16);
            const float lse_a = am + logf(sa);
            // lse_b = logsumexp(logpi - tau*z)
            float bv0 = lp[0] - TAU * zf[0][v], bv1 = lp[1] - TAU * zf[1][v];
            float bv2 = lp[2] - TAU * zf[2][v], bv3 = lp[3] - TAU * zf[3][v];
            float bm = fmaxf(fmaxf(bv0, bv1), fmaxf(bv2, bv3));
            for (int m = 1; m < 16; m <<= 1) bm = fmaxf(bm, __shfl_xor(bm, m, 16));
            float sb = expf(bv0 - bm) + expf(bv1 - bm) + expf(bv2 - bm) + expf(bv3 - bm);
            for (int m = 1; m < 16; m <<= 1) sb += __shfl_xor(sb, m, 16);
            const float lse_b = bm + logf(sb);

            const float mix     = lse_a - lse_z;
            const float sumlogz = sz - (float)KDIM * lse_z;
            const float lset    = TAU * lse_z + lse_b;
            const float con     = const0 - (TAU + 1.0f) * sumlogz - (float)KDIM * lset;
            wsum += (l16 == 0) ? (con + mix) : 0.f;   // lanes 0 & 16: rows v, v+8
        }
    }
    for (int m = 1; m < 32; m <<= 1) wsum += __shfl_xor(wsum, m, 32);
    if (lane == 0) wavePart[wgid] = -wsum;   // z_loss = -(con+mix).sum()
}

// ---------------------------------------------------------------------------
// Kernel D: deterministic final reduction
// ---------------------------------------------------------------------------
__global__ void kfinal(const float* __restrict__ wavePart,
                       const float* __restrict__ wsScal, float* __restrict__ out) {
    __shared__ float s[256];
    const int t = threadIdx.x;
    float a = 0.f;
    for (int i = t; i < TOTWAVES; i += 256) a += wavePart[i];
    s[t] = a; __syncthreads();
    for (int off = 128; off > 0; off >>= 1) {
        if (t < off) s[t] += s[t + off];
        __syncthreads();
    }
    if (t == 0) out[0] = s[0] + wsScal[1];
}

// ---------------------------------------------------------------------------
extern "C" void kernel_launch(void* const* d_in, const int* in_sizes, int n_in,
                              void* d_out, int out_size, void* d_ws, size_t ws_size,
                              hipStream_t stream) {
    const float* met       = (const float*)d_in[0];
    const float* mu        = (const float*)d_in[1];
    const float* pi        = (const float*)d_in[2];
    const float* lambda_mu = (const float*)d_in[3];
    const float* b         = (const float*)d_in[4];
    const float* Cv        = (const float*)d_in[5];
    const float* r         = (const float*)d_in[6];
    const float* z         = (const float*)d_in[7];
    float* out = (float*)d_out;

    float* w = (float*)d_ws;
    float*  wsScal   = w;                         // 16 floats reserved
    float4* wsK      = (float4*)(w + 16);         // 64 float4
    float*  pmin     = w + 16 + 4 * KDIM;         // AGRID*16
    float*  pmax     = pmin + AGRID * DDIM;       // AGRID*16
    float*  wavePart = pmax + AGRID * DDIM;       // TOTWAVES floats

    kminmax<<<AGRID, ABLK, 0, stream>>>(met, pmin, pmax);
    kprep<<<1, KDIM, 0, stream>>>(mu, pi, lambda_mu, b, Cv, r, pmin, pmax, wsScal, wsK);
    kmain<<<CGRID, CBLK, 0, stream>>>(met, mu, z, wsScal, wsK, wavePart);
    kfinal<<<1, 256, 0, stream>>>(wavePart, wsScal, out);
}